// CCL_67723044323788
// MI455X (gfx1250) — compile-verified
//
#include <hip/hip_runtime.h>
#include <math.h>

typedef __attribute__((ext_vector_type(2))) float v2f;
typedef __attribute__((ext_vector_type(8))) float v8f;

#define NROWS 8192
#define DIM   128
#define NCLS  1000
#define INV_T 10.0f
#define EPSL  1e-10f

#define TPITCH 132                 // padded LDS row pitch (floats): conflict-free b64 reads
#define TILEF  (16 * TPITCH)       // floats per staged 16-col tile

// ---------------- utility kernels ----------------

__global__ void zero_kernel(float* p, int n) {
  int i = blockIdx.x * blockDim.x + threadIdx.x;
  if (i < n) p[i] = 0.0f;
}

// One wave per row: 32 lanes x float4 = 128 elems. L2-normalize (eps=1e-12).
__global__ __launch_bounds__(256) void normalize_kernel(const float* __restrict__ in,
                                                        float* __restrict__ out) {
  int wave = threadIdx.x >> 5;
  int lane = threadIdx.x & 31;
  int row  = blockIdx.x * 8 + wave;
  const float4* src = (const float4*)(in + (size_t)row * DIM);
  float4 v = src[lane];
  float ss = v.x * v.x + v.y * v.y + v.z * v.z + v.w * v.w;
  #pragma unroll
  for (int m = 16; m >= 1; m >>= 1) ss += __shfl_xor(ss, m, 32);
  float inv = 1.0f / fmaxf(sqrtf(ss), 1e-12f);
  float4 o;
  o.x = v.x * inv; o.y = v.y * inv; o.z = v.z * inv; o.w = v.w * inv;
  ((float4*)(out + (size_t)row * DIM))[lane] = o;
}

__global__ void hist_kernel(const int* __restrict__ tgt, float* __restrict__ cnt) {
  int i = blockIdx.x * blockDim.x + threadIdx.x;
  if (i < NROWS) atomicAdd(&cnt[tgt[i]], 1.0f);
}

// Stage a pair of 16-col B tiles (cols [c0,c0+32)) into LDS buffer `buf`
// via gfx1250 async global->LDS (tracked by ASYNCcnt). 128 threads issue
// 8 x b128 each; caller later does s_wait_asynccnt 0 + barrier.
__device__ __forceinline__ void stage_pair(const float* __restrict__ f2n,
                                           float* sbase, int c0, int buf, int tid) {
  #pragma unroll
  for (int k = 0; k < 4; ++k) {
    int idx = tid + k * 128;     // 0..511
    int row = idx >> 5;          // 0..15 within tile
    int c4  = idx & 31;          // float4 column chunk
    #pragma unroll
    for (int tt = 0; tt < 2; ++tt) {
      const float* g = f2n + (size_t)(c0 + tt * 16 + row) * DIM + c4 * 4;
      unsigned lo = (unsigned)(size_t)(sbase + (buf * 2 + tt) * TILEF + row * TPITCH + c4 * 4);
      asm volatile("global_load_async_to_lds_b128 %0, %1, off"
                   :: "v"(lo), "v"(g) : "memory");
    }
  }
}

__device__ __forceinline__ void wait_async_all() {
  asm volatile("s_wait_asynccnt 0" ::: "memory");
}

// ---------------- NCE pass 1: row sums of exp(cos/T) ----------------
// grid (128 rowgroups, 16 colslices), block = 128 (4 waves).
// Each wave owns a 16-row A panel (64 rows/block); all waves share B tiles
// staged through LDS (double-buffered async copies), 2 WMMA chains per wave.
__global__ __launch_bounds__(128) void nce_pass1(const float* __restrict__ f1n,
                                                 const float* __restrict__ f2n,
                                                 float* __restrict__ S) {
  __shared__ __align__(16) float sbuf[4 * TILEF];   // 2 buffers x 2 tiles

  const int tid  = threadIdx.x;
  const int w    = tid >> 5;
  const int lane = tid & 31;
  const int li   = lane & 15;
  const int hf   = lane >> 4;
  const int rb16 = blockIdx.x * 64 + w * 16;        // this wave's 16 rows
  const int colbase = blockIdx.y * 512;             // 32 tiles of 16 cols

  // A panel: 16 rows x 128 K, ISA layout for 16x4 f32 A tiles, preloaded.
  const float* arow = f1n + (size_t)(rb16 + li) * DIM + 2 * hf;
  v2f a[32];
  #pragma unroll
  for (int s = 0; s < 32; ++s) a[s] = *(const v2f*)(arow + 4 * s);

  float rsum[8];
  #pragma unroll
  for (int k = 0; k < 8; ++k) rsum[k] = 0.0f;

  stage_pair(f2n, sbuf, colbase, 0, tid);
  wait_async_all();
  __syncthreads();

  for (int p = 0; p < 16; ++p) {                    // 16 tile-pairs
    if (p < 15) stage_pair(f2n, sbuf, colbase + (p + 1) * 32, (p + 1) & 1, tid);

    const float* b0p = &sbuf[((p & 1) * 2 + 0) * TILEF + li * TPITCH + 2 * hf];
    const float* b1p = &sbuf[((p & 1) * 2 + 1) * TILEF + li * TPITCH + 2 * hf];
    v8f acc0 = {}, acc1 = {};
    #pragma unroll
    for (int s = 0; s < 32; ++s) {
      v2f b0 = *(const v2f*)(b0p + 4 * s);
      v2f b1 = *(const v2f*)(b1p + 4 * s);
      acc0 = __builtin_amdgcn_wmma_f32_16x16x4_f32(false, a[s], false, b0, (short)0, acc0, false, false);
      acc1 = __builtin_amdgcn_wmma_f32_16x16x4_f32(false, a[s], false, b1, (short)0, acc1, false, false);
    }
    #pragma unroll
    for (int k = 0; k < 8; ++k)
      rsum[k] += __expf(acc0[k] * INV_T) + __expf(acc1[k] * INV_T);

    wait_async_all();
    __syncthreads();
  }

  // reduce over the 16 lanes of each half (columns), atomically add per row
  #pragma unroll
  for (int k = 0; k < 8; ++k) {
    float v = rsum[k];
    v += __shfl_xor(v, 8, 16);
    v += __shfl_xor(v, 4, 16);
    v += __shfl_xor(v, 2, 16);
    v += __shfl_xor(v, 1, 16);
    if (li == 0) atomicAdd(&S[rb16 + k + 8 * hf], v);
  }
}

// ---------------- NCE pass 2: masked log terms ----------------
__global__ __launch_bounds__(128) void nce_pass2(const float* __restrict__ f1n,
                                                 const float* __restrict__ f2n,
                                                 const float* __restrict__ S,
                                                 const int* __restrict__ tgt,
                                                 float* __restrict__ Lpos,
                                                 float* __restrict__ Lneg) {
  __shared__ __align__(16) float sbuf[4 * TILEF];

  const int tid  = threadIdx.x;
  const int w    = tid >> 5;
  const int lane = tid & 31;
  const int li   = lane & 15;
  const int hf   = lane >> 4;
  const int rb16 = blockIdx.x * 64 + w * 16;
  const int colbase = blockIdx.y * 512;

  const float* arow = f1n + (size_t)(rb16 + li) * DIM + 2 * hf;
  v2f a[32];
  #pragma unroll
  for (int s = 0; s < 32; ++s) a[s] = *(const v2f*)(arow + 4 * s);

  float invS[8];
  int tr[8];
  #pragma unroll
  for (int k = 0; k < 8; ++k) {
    int r = rb16 + k + 8 * hf;
    invS[k] = 1.0f / S[r];
    tr[k]   = tgt[r];
  }

  float lp[8], ln[8];
  #pragma unroll
  for (int k = 0; k < 8; ++k) { lp[k] = 0.0f; ln[k] = 0.0f; }

  stage_pair(f2n, sbuf, colbase, 0, tid);
  wait_async_all();
  __syncthreads();

  for (int p = 0; p < 16; ++p) {
    if (p < 15) stage_pair(f2n, sbuf, colbase + (p + 1) * 32, (p + 1) & 1, tid);

    int c0  = colbase + p * 32;
    int tc0 = tgt[c0 + li];
    int tc1 = tgt[c0 + 16 + li];
    const float* b0p = &sbuf[((p & 1) * 2 + 0) * TILEF + li * TPITCH + 2 * hf];
    const float* b1p = &sbuf[((p & 1) * 2 + 1) * TILEF + li * TPITCH + 2 * hf];
    v8f acc0 = {}, acc1 = {};
    #pragma unroll
    for (int s = 0; s < 32; ++s) {
      v2f b0 = *(const v2f*)(b0p + 4 * s);
      v2f b1 = *(const v2f*)(b1p + 4 * s);
      acc0 = __builtin_amdgcn_wmma_f32_16x16x4_f32(false, a[s], false, b0, (short)0, acc0, false, false);
      acc1 = __builtin_amdgcn_wmma_f32_16x16x4_f32(false, a[s], false, b1, (short)0, acc1, false, false);
    }
    #pragma unroll
    for (int k = 0; k < 8; ++k) {
      float ps0 = __expf(acc0[k] * INV_T) * invS[k];
      float ps1 = __expf(acc1[k] * INV_T) * invS[k];
      bool p0 = (tc0 == tr[k]);
      bool p1 = (tc1 == tr[k]);
      lp[k] += (p0 ? -__logf(ps0 + EPSL) : 0.0f) +
               (p1 ? -__logf(ps1 + EPSL) : 0.0f);
      ln[k] += (p0 ? 0.0f : -__logf(1.0f - ps0 + EPSL)) +
               (p1 ? 0.0f : -__logf(1.0f - ps1 + EPSL));
    }

    wait_async_all();
    __syncthreads();
  }

  #pragma unroll
  for (int k = 0; k < 8; ++k) {
    float vp = lp[k], vn = ln[k];
    vp += __shfl_xor(vp, 8, 16); vn += __shfl_xor(vn, 8, 16);
    vp += __shfl_xor(vp, 4, 16); vn += __shfl_xor(vn, 4, 16);
    vp += __shfl_xor(vp, 2, 16); vn += __shfl_xor(vn, 2, 16);
    vp += __shfl_xor(vp, 1, 16); vn += __shfl_xor(vn, 1, 16);
    if (li == 0) {
      int r = rb16 + k + 8 * hf;
      atomicAdd(&Lpos[r], vp);
      atomicAdd(&Lneg[r], vn);
    }
  }
}

// ---------------- JSD ----------------
__device__ __forceinline__ float block_reduce_sum(float v, float* sm) {
  int tid = threadIdx.x;
  sm[tid] = v; __syncthreads();
  for (int o = 128; o > 0; o >>= 1) {
    if (tid < o) sm[tid] += sm[tid + o];
    __syncthreads();
  }
  float r = sm[0]; __syncthreads();
  return r;
}

__device__ __forceinline__ float block_reduce_max(float v, float* sm) {
  int tid = threadIdx.x;
  sm[tid] = v; __syncthreads();
  for (int o = 128; o > 0; o >>= 1) {
    if (tid < o) sm[tid] = fmaxf(sm[tid], sm[tid + o]);
    __syncthreads();
  }
  float r = sm[0]; __syncthreads();
  return r;
}

// One block (256 threads) per row of the 8192x1000 logit matrices.
__global__ __launch_bounds__(256) void jsd_kernel(const float* __restrict__ ls,
                                                  const float* __restrict__ lt,
                                                  float* __restrict__ accum) {
  __shared__ float sm[256];
  int row = blockIdx.x;
  int tid = threadIdx.x;
  const float* rs = ls + (size_t)row * NCLS;
  const float* rt = lt + (size_t)row * NCLS;

  float xs[4], xt[4];
  float mxs = -INFINITY, mxt = -INFINITY;
  #pragma unroll
  for (int k = 0; k < 4; ++k) {
    int j = tid + k * 256;
    if (j < NCLS) {
      xs[k] = rs[j]; xt[k] = rt[j];
      mxs = fmaxf(mxs, xs[k]); mxt = fmaxf(mxt, xt[k]);
    } else {
      xs[k] = -INFINITY; xt[k] = -INFINITY;
    }
  }
  mxs = block_reduce_max(mxs, sm);
  mxt = block_reduce_max(mxt, sm);

  float ses = 0.0f, set = 0.0f;
  #pragma unroll
  for (int k = 0; k < 4; ++k) {
    ses += __expf(xs[k] - mxs);   // exp(-inf) = 0 for padding
    set += __expf(xt[k] - mxt);
  }
  ses = block_reduce_sum(ses, sm);
  set = block_reduce_sum(set, sm);
  float logZs = mxs + __logf(ses);
  float logZt = mxt + __logf(set);

  // kl_st + kl_ts = sum (pt - ps) * (log pt - log ps)
  float c = 0.0f;
  #pragma unroll
  for (int k = 0; k < 4; ++k) {
    int j = tid + k * 256;
    if (j < NCLS) {
      float lps = xs[k] - logZs;
      float lpt = xt[k] - logZt;
      float d = lpt - lps;
      c += (__expf(lpt) - __expf(lps)) * d;
    }
  }
  c = block_reduce_sum(c, sm);
  if (tid == 0) atomicAdd(&accum[1], c * (0.5f / (float)NROWS));
}

// ---------------- finalize ----------------
__global__ __launch_bounds__(256) void nce_finalize(const float* __restrict__ Lpos,
                                                    const float* __restrict__ Lneg,
                                                    const float* __restrict__ cnt,
                                                    const int* __restrict__ tgt,
                                                    float* __restrict__ accum) {
  __shared__ float sm[256];
  int i = blockIdx.x * 256 + threadIdx.x;
  float c    = cnt[tgt[i]];
  float term = Lpos[i] / c + Lneg[i] / ((float)NROWS - c);
  float tot  = block_reduce_sum(term, sm);
  if (threadIdx.x == 0) atomicAdd(&accum[0], tot);
}

__global__ void write_out(const float* __restrict__ accum, float* __restrict__ out) {
  out[0] = accum[0] * (1.0f / (float)NROWS) + accum[1];
}

// ---------------- launch ----------------
extern "C" void kernel_launch(void* const* d_in, const int* in_sizes, int n_in,
                              void* d_out, int out_size, void* d_ws, size_t ws_size,
                              hipStream_t stream) {
  const float* fs = (const float*)d_in[0];
  const float* ft = (const float*)d_in[1];
  const float* ls = (const float*)d_in[2];
  const float* lt = (const float*)d_in[3];
  const int*  tgt = (const int*)d_in[4];
  float* out = (float*)d_out;

  float* wsf  = (float*)d_ws;
  float* f1n  = wsf;                       // 8192*128
  float* f2n  = f1n + NROWS * DIM;         // 8192*128
  float* S    = f2n + NROWS * DIM;         // 8192
  float* Lpos = S + NROWS;                 // 8192
  float* Lneg = Lpos + NROWS;              // 8192
  float* cnt  = Lneg + NROWS;              // 128 (targets < 100)
  float* accum = cnt + 128;                // [0]=nce sum, [1]=jsd

  int nz = NROWS * 3 + 128 + 2;
  zero_kernel<<<(nz + 255) / 256, 256, 0, stream>>>(S, nz);

  normalize_kernel<<<NROWS / 8, 256, 0, stream>>>(fs, f1n);
  normalize_kernel<<<NROWS / 8, 256, 0, stream>>>(ft, f2n);
  hist_kernel<<<NROWS / 256, 256, 0, stream>>>(tgt, cnt);

  dim3 g(NROWS / 64, 16);   // 128 row-groups (64 rows each) x 16 column slices
  nce_pass1<<<g, 128, 0, stream>>>(f1n, f2n, S);
  nce_pass2<<<g, 128, 0, stream>>>(f1n, f2n, S, tgt, Lpos, Lneg);

  jsd_kernel<<<NROWS, 256, 0, stream>>>(ls, lt, accum);
  nce_finalize<<<NROWS / 256, 256, 0, stream>>>(Lpos, Lneg, cnt, tgt, accum);
  write_out<<<1, 1, 0, stream>>>(accum, out);
}